// RGCNConv_22239340658867
// MI455X (gfx1250) — compile-verified
//
#include <hip/hip_runtime.h>

typedef float v2f __attribute__((ext_vector_type(2)));
typedef float v8f __attribute__((ext_vector_type(8)));

#define D_DIM 128
#define TILE_M 16
#define LDS_STRIDE 132   // 128 + 4 pad -> lane m hits banks 4m.. (conflict-free ds_load_b64)

// ---------------- bucketing kernels (counting sort of edges by relation) ----------------

__global__ void k_init_counts(int* cnt, int* cursor, int R) {
    int i = threadIdx.x;
    if (i < R) { cnt[i] = 0; cursor[i] = 0; }
}

__global__ void k_hist(const int* __restrict__ etype, int* cnt, int E) {
    int e = blockIdx.x * 256 + threadIdx.x;
    if (e < E) atomicAdd(&cnt[etype[e]], 1);
}

// single block: exclusive scan of tile-padded bucket sizes + fill tile->relation map
__global__ void k_scan_tiles(const int* __restrict__ cnt, int* offPadEdges,
                             int* tileRel, int R, int maxTiles) {
    __shared__ int offT[129];
    if (threadIdx.x == 0) {
        int acc = 0;
        for (int r = 0; r < R; ++r) {
            offT[r] = acc;
            acc += (cnt[r] + TILE_M - 1) / TILE_M;
        }
        offT[R] = acc;
    }
    __syncthreads();
    if ((int)threadIdx.x <= R) offPadEdges[threadIdx.x] = offT[threadIdx.x] * TILE_M;
    for (int t = threadIdx.x; t < maxTiles; t += blockDim.x) tileRel[t] = -1;
    __syncthreads();
    for (int r = 0; r < R; ++r)
        for (int t = offT[r] + threadIdx.x; t < offT[r + 1]; t += blockDim.x)
            tileRel[t] = r;
}

__global__ void k_fill_sorted(int* sorted, int len) {
    int i = blockIdx.x * 256 + threadIdx.x;
    if (i < len) sorted[i] = -1;
}

__global__ void k_scatter(const int* __restrict__ etype, const int* __restrict__ offPadEdges,
                          int* cursor, int* sorted, int E) {
    int e = blockIdx.x * 256 + threadIdx.x;
    if (e < E) {
        int r = etype[e];
        int pos = atomicAdd(&cursor[r], 1);
        sorted[offPadEdges[r] + pos] = e;
    }
}

// ---------------- WMMA GEMM kernels ----------------
// Block = 256 threads = 8 waves. Block computes a 16x128 C tile; wave w owns cols [16w,16w+16).
// K = 128 swept by 32 x V_WMMA_F32_16X16X4_F32 per wave.
// A frag (16x4 f32): lane m=lane&15, half=lane>>4 -> a = A[m][k0+2h .. k0+2h+1]
// B frag (4x16 f32): b = B[k0+2h .. k0+2h+1][col]
// C frag (16x16 f32): c[v] = C[v + 8h][col]

// out = x @ root  (initializes the output)
__global__ void __launch_bounds__(256, 2)
k_root_gemm(const float* __restrict__ x, const float* __restrict__ root,
            float* __restrict__ out, int N) {
    __shared__ float As[TILE_M * LDS_STRIDE];
    const int tid  = threadIdx.x;
    const int base = blockIdx.x * TILE_M;

    {   // stage A tile: 16 rows x 128 cols, 8 floats per thread
        int mrow = tid >> 4;
        int colb = (tid & 15) * 8;
        int grow = base + mrow;
        float4 a0 = {0, 0, 0, 0}, a1 = {0, 0, 0, 0};
        if (grow < N) {
            const float4* p = (const float4*)(x + grow * D_DIM + colb);
            a0 = p[0]; a1 = p[1];
        }
        float* dst = &As[mrow * LDS_STRIDE + colb];
        dst[0] = a0.x; dst[1] = a0.y; dst[2] = a0.z; dst[3] = a0.w;
        dst[4] = a1.x; dst[5] = a1.y; dst[6] = a1.z; dst[7] = a1.w;
    }

    const int wave = tid >> 5, lane = tid & 31;
    const int half = lane >> 4, nloc = lane & 15;
    const int col  = wave * 16 + nloc;

    v2f breg[32];
#pragma unroll
    for (int s = 0; s < 32; ++s) {
        int kk = 4 * s + 2 * half;
        breg[s].x = root[kk * D_DIM + col];
        breg[s].y = root[(kk + 1) * D_DIM + col];
    }
    __syncthreads();

    v8f c = {};
    const int m = lane & 15;
#pragma unroll
    for (int s = 0; s < 32; ++s) {
        int kk = 4 * s + 2 * half;
        v2f a;
        a.x = As[m * LDS_STRIDE + kk];
        a.y = As[m * LDS_STRIDE + kk + 1];
        c = __builtin_amdgcn_wmma_f32_16x16x4_f32(false, a, false, breg[s],
                                                  (short)0, c, false, false);
    }
#pragma unroll
    for (int v = 0; v < 8; ++v) {
        int row = base + v + 8 * half;
        if (row < N) out[row * D_DIM + col] = c[v];
    }
}

// per 16-edge tile (single relation): C = (norm * x[src]) @ weight[r]; atomic scatter to out[src]
__global__ void __launch_bounds__(256, 2)
k_edge_gemm(const float* __restrict__ x, const float* __restrict__ weight,
            const int* __restrict__ eidx,     // row 0 of edge_index (src), length E
            const float* __restrict__ enorm,
            const int* __restrict__ sorted,
            const int* __restrict__ tileRel,
            float* __restrict__ out) {
    __shared__ float As[TILE_M * LDS_STRIDE];
    __shared__ int   nodeIds[TILE_M];

    const int t = blockIdx.x;
    const int r = tileRel[t];
    if (r < 0) return;                 // uniform per block -> EXEC stays all-ones at WMMA
    const int tid = threadIdx.x;

    {   // stage A tile with gather + edge_norm folded in
        int mrow = tid >> 4;
        int colb = (tid & 15) * 8;
        int eid  = sorted[t * TILE_M + mrow];
        float4 a0 = {0, 0, 0, 0}, a1 = {0, 0, 0, 0};
        int   node = -1;
        float nrm  = 0.0f;
        if (eid >= 0) {
            node = eidx[eid];
            nrm  = enorm[eid];
            const float4* p = (const float4*)(x + node * D_DIM + colb);
            a0 = p[0]; a1 = p[1];
        }
        if ((tid & 15) == 0) nodeIds[mrow] = node;
        float* dst = &As[mrow * LDS_STRIDE + colb];
        dst[0] = a0.x * nrm; dst[1] = a0.y * nrm; dst[2] = a0.z * nrm; dst[3] = a0.w * nrm;
        dst[4] = a1.x * nrm; dst[5] = a1.y * nrm; dst[6] = a1.z * nrm; dst[7] = a1.w * nrm;
    }

    const int wave = tid >> 5, lane = tid & 31;
    const int half = lane >> 4, nloc = lane & 15;
    const int col  = wave * 16 + nloc;
    const float* W = weight + r * (D_DIM * D_DIM);

    v2f breg[32];
#pragma unroll
    for (int s = 0; s < 32; ++s) {
        int kk = 4 * s + 2 * half;
        breg[s].x = W[kk * D_DIM + col];
        breg[s].y = W[(kk + 1) * D_DIM + col];
    }
    __syncthreads();

    v8f c = {};
    const int m = lane & 15;
#pragma unroll
    for (int s = 0; s < 32; ++s) {
        int kk = 4 * s + 2 * half;
        v2f a;
        a.x = As[m * LDS_STRIDE + kk];
        a.y = As[m * LDS_STRIDE + kk + 1];
        c = __builtin_amdgcn_wmma_f32_16x16x4_f32(false, a, false, breg[s],
                                                  (short)0, c, false, false);
    }
#pragma unroll
    for (int v = 0; v < 8; ++v) {
        int mrow = v + 8 * half;
        int node = nodeIds[mrow];
        if (node >= 0) atomicAdd(&out[node * D_DIM + col], c[v]);
    }
}

// ---------------- launch ----------------

extern "C" void kernel_launch(void* const* d_in, const int* in_sizes, int n_in,
                              void* d_out, int out_size, void* d_ws, size_t ws_size,
                              hipStream_t stream) {
    const float* x      = (const float*)d_in[0];
    const float* weight = (const float*)d_in[1];
    const float* root   = (const float*)d_in[2];
    const int*   eindex = (const int*)d_in[3];   // [2,E] flat; row 0 = src
    const int*   etype  = (const int*)d_in[4];
    const float* enorm  = (const float*)d_in[5];
    float* out = (float*)d_out;

    const int N = in_sizes[0] / D_DIM;
    const int R = in_sizes[1] / (D_DIM * D_DIM);
    const int E = in_sizes[4];

    const int maxTiles  = (E + TILE_M - 1) / TILE_M + R;  // tile-padded worst case
    const int sortedLen = maxTiles * TILE_M;

    // workspace layout (all int32)
    char* ws = (char*)d_ws;
    int* cnt        = (int*)(ws + 0);          // 128 ints
    int* cursor     = (int*)(ws + 512);        // 128 ints
    int* offPad     = (int*)(ws + 1024);       // R+1 ints (<=129)
    int* tileRel    = (int*)(ws + 2048);       // maxTiles ints
    size_t tr_bytes = ((size_t)maxTiles * 4 + 255) & ~(size_t)255;
    int* sorted     = (int*)(ws + 2048 + tr_bytes);   // sortedLen ints

    k_init_counts<<<1, 256, 0, stream>>>(cnt, cursor, R);
    k_hist<<<(E + 255) / 256, 256, 0, stream>>>(etype, cnt, E);
    k_scan_tiles<<<1, 256, 0, stream>>>(cnt, offPad, tileRel, R, maxTiles);
    k_fill_sorted<<<(sortedLen + 255) / 256, 256, 0, stream>>>(sorted, sortedLen);
    k_scatter<<<(E + 255) / 256, 256, 0, stream>>>(etype, offPad, cursor, sorted, E);

    k_root_gemm<<<(N + TILE_M - 1) / TILE_M, 256, 0, stream>>>(x, root, out, N);
    k_edge_gemm<<<maxTiles, 256, 0, stream>>>(x, weight, eindex, enorm, sorted, tileRel, out);
}